// semantic_68212670595850
// MI455X (gfx1250) — compile-verified
//
#include <hip/hip_runtime.h>
#include <math.h>

typedef __attribute__((ext_vector_type(2))) float v2f;
typedef __attribute__((ext_vector_type(4))) float v4f;
typedef __attribute__((ext_vector_type(8))) float v8f;

#define D_IMG 2048
#define D_WORD 300
#define D_MID 1024
#define NCLS 80
#define HW 196
#define NB 8
#define NP (NB * HW)   // 1568

#define APAD 34        // A row stride (32+2): conflict-free, 8B aligned
#define BPAD 36        // B row stride (32+4): conflict-free, 16B aligned rows

__device__ __forceinline__ v8f wmma_f32_4(v2f a, v2f b, v8f c) {
  // V_WMMA_F32_16X16X4_F32 : D = A(16x4,f32) * B(4x16,f32) + C(16x16,f32)
  return __builtin_amdgcn_wmma_f32_16x16x4_f32(
      /*neg_a=*/false, a, /*neg_b=*/false, b,
      /*c_mod=*/(short)0, c, /*reuse_a=*/false, /*reuse_b=*/false);
}

// gfx1250 has hardware V_TANH_F32 (TRANS pipe); use it if clang exposes it.
#if __has_builtin(__builtin_amdgcn_tanhf)
__device__ __forceinline__ float fast_tanh(float x) { return __builtin_amdgcn_tanhf(x); }
#else
#warning "CDNA5 probe: __builtin_amdgcn_tanhf NOT available - falling back to libm tanhf"
__device__ __forceinline__ float fast_tanh(float x) { return tanhf(x); }
#endif

// ---------------------------------------------------------------------------
// Kernel 1: f_wd[c,m] = sum_d word[c,d] * W2[m,d]      (80 x 1024, K=300)
// ---------------------------------------------------------------------------
__global__ void k_fwd(const float* __restrict__ word,
                      const float* __restrict__ W2,
                      float* __restrict__ fwd_) {
  int idx = blockIdx.x * blockDim.x + threadIdx.x;
  if (idx >= NCLS * D_MID) return;
  int c = idx / D_MID, m = idx - c * D_MID;
  const float* wr  = word + (size_t)c * D_WORD;
  const float* w2r = W2   + (size_t)m * D_WORD;
  float s = 0.f;
  for (int d = 0; d < D_WORD; ++d) s = fmaf(wr[d], w2r[d], s);
  fwd_[idx] = s;
}

// ---------------------------------------------------------------------------
// Kernel 2: v[m] = sum_n Wa[n]*W3[n,m] ;  c0 = sum_n b3[n]*Wa[n] + ba
// ---------------------------------------------------------------------------
__global__ void k_vc(const float* __restrict__ W3, const float* __restrict__ Wa,
                     const float* __restrict__ b3, const float* __restrict__ ba,
                     float* __restrict__ v, float* __restrict__ c0) {
  if (blockIdx.x < 4) {
    int m = blockIdx.x * 256 + threadIdx.x;
    float s = 0.f;
    for (int n = 0; n < D_MID; ++n) s = fmaf(Wa[n], W3[(size_t)n * D_MID + m], s);
    v[m] = s;
  } else {
    __shared__ float red[256];
    float s = 0.f;
    for (int n = threadIdx.x; n < D_MID; n += 256) s = fmaf(b3[n], Wa[n], s);
    red[threadIdx.x] = s;
    __syncthreads();
    for (int off = 128; off > 0; off >>= 1) {
      if (threadIdx.x < off) red[threadIdx.x] += red[threadIdx.x + off];
      __syncthreads();
    }
    if (threadIdx.x == 0) c0[0] = red[0] + ba[0];
  }
}

// ---------------------------------------------------------------------------
// Kernel 3 (WMMA): f_wh[p,m] = sum_d A[p,d]*W1[m,d]
//   A[p,d] = img[b, d, hw]  (p = b*196 + hw) -> K-major in memory.
//   Macro-tile 64(p) x 32(m), 256 threads = 8 waves -> 4x2 grid of 16x16
//   tiles; K staged through LDS in chunks of 32.
//   Out-of-range A rows are CLAMPED, not zeroed: WMMA row m only feeds output
//   row m, and those rows are skipped at store time -> branchless staging.
//   Epilogue: uniform fast path for full tiles (only blockIdx.y==24 is ragged).
// ---------------------------------------------------------------------------
__global__ __launch_bounds__(256) void k_gemm_fwh(const float* __restrict__ img,
                                                  const float* __restrict__ W1,
                                                  float* __restrict__ fwh) {
  __shared__ float sA[64 * APAD];
  __shared__ float sB[32 * BPAD];
  const int tid   = threadIdx.x;
  const int lane  = tid & 31;
  const int wid   = tid >> 5;
  const int lmod  = lane & 15;
  const int lhalf = lane >> 4;          // lanes 16-31 hold K+2 per ISA layout
  const int ra = (wid & 3) * 16;        // wave row-tile base within macro-tile
  const int cb = (wid >> 2) * 16;       // wave col-tile base
  const int p0 = blockIdx.y * 64;
  const int m0 = blockIdx.x * 32;

  // ---- A-tile staging: thread always fills row r, k = kb + 4i (hoisted)
  const int r  = tid & 63;
  const int kb = tid >> 6;              // 0..3
  int p = p0 + r;
  if (p >= NP) p = NP - 1;              // clamp: dead rows, never stored
  const int bA = p / HW, hwA = p - (p / HW) * HW;
  const float* aSrc = img + (size_t)bA * D_IMG * HW + hwA;
  float* sArow = sA + r * APAD;

  // ---- B-tile staging: thread fills row n = tid>>3, 4 consecutive k
  const int bn = tid >> 3;              // 0..31
  const int bk = (tid & 7) * 4;         // 0,4,...,28
  const float* bSrc = W1 + (size_t)(m0 + bn) * D_IMG + bk;
  float* sBdst = sB + bn * BPAD + bk;   // 16B aligned (BPAD=36)

  v8f acc = {};
  for (int k0 = 0; k0 < D_IMG; k0 += 32) {
#pragma unroll
    for (int i = 0; i < 8; ++i) {
      const int k = kb + 4 * i;
      sArow[k] = aSrc[(size_t)(k0 + k) * HW];
    }
    *(v4f*)sBdst = *(const v4f*)(bSrc + k0);
    __syncthreads();
#pragma unroll
    for (int kk = 0; kk < 32; kk += 4) {
      const int ks = kk + 2 * lhalf;
      v2f a, b;
      a.x = sA[(ra + lmod) * APAD + ks];
      a.y = sA[(ra + lmod) * APAD + ks + 1];
      b.x = sB[(cb + lmod) * BPAD + ks];
      b.y = sB[(cb + lmod) * BPAD + ks + 1];
      acc = wmma_f32_4(a, b, acc);
    }
    __syncthreads();
  }
  // C/D layout: lanes 0-15 rows i, lanes 16-31 rows i+8; col = lane%16
  const int prBase = p0 + ra + 8 * lhalf;
  float* dst = fwh + (size_t)prBase * D_MID + (m0 + cb + lmod);
  if (p0 + 64 <= NP) {                  // wave-uniform fast path
#pragma unroll
    for (int i = 0; i < 8; ++i) dst[(size_t)i * D_MID] = acc[i];
  } else {
#pragma unroll
    for (int i = 0; i < 8; ++i)
      if (prBase + i < NP) dst[(size_t)i * D_MID] = acc[i];
  }
}

// ---------------------------------------------------------------------------
// Kernel 4: coef[p,c] = c0 + sum_m tanh(f_wh[p,m]*f_wd[c,m]) * v[m]
//   One block per spatial position p; tanh-bound (128M trans ops total).
// ---------------------------------------------------------------------------
__global__ __launch_bounds__(256) void k_coef(const float* __restrict__ fwh,
                                              const float* __restrict__ fwd_,
                                              const float* __restrict__ v,
                                              const float* __restrict__ c0p,
                                              float* __restrict__ coef) {
  __shared__ float sF[D_MID];
  __shared__ float sV[D_MID];
  __shared__ float red[8];
  const int p = blockIdx.x;
  const int tid = threadIdx.x;
  for (int j = tid; j < D_MID; j += 256) {
    sF[j] = fwh[(size_t)p * D_MID + j];
    sV[j] = v[j];
  }
  __syncthreads();
  const float c0 = c0p[0];
  const int lane = tid & 31, wid = tid >> 5;
  for (int c = 0; c < NCLS; ++c) {
    const float* fr = fwd_ + (size_t)c * D_MID;
    float s = 0.f;
#pragma unroll
    for (int u = 0; u < 4; ++u) {
      const int j = tid + u * 256;
      s = fmaf(fast_tanh(sF[j] * fr[j]), sV[j], s);
    }
    // wave32 reduce
    for (int off = 16; off > 0; off >>= 1)
      s += __shfl_down(s, off, 32);
    if (lane == 0) red[wid] = s;
    __syncthreads();
    if (tid == 0) {
      float t = 0.f;
      for (int w = 0; w < 8; ++w) t += red[w];
      coef[(size_t)p * NCLS + c] = t + c0;
    }
    __syncthreads();
  }
}

// ---------------------------------------------------------------------------
// Kernel 5: softmax over 196 spatial positions per (b,c); in-place on coef.
// ---------------------------------------------------------------------------
__global__ __launch_bounds__(256) void k_softmax(float* __restrict__ coef) {
  __shared__ float sh[256];
  const int bc = blockIdx.x;            // b*NCLS + c
  const int b = bc / NCLS, c = bc - b * NCLS;
  const int t = threadIdx.x;
  const size_t base = (size_t)b * HW * NCLS + c;
  float val = -3.0e38f;
  if (t < HW) val = coef[base + (size_t)t * NCLS];
  sh[t] = val;
  __syncthreads();
  for (int off = 128; off > 0; off >>= 1) {
    if (t < off) sh[t] = fmaxf(sh[t], sh[t + off]);
    __syncthreads();
  }
  const float mx = sh[0];
  __syncthreads();
  float ex = (t < HW) ? __expf(val - mx) : 0.f;
  sh[t] = ex;
  __syncthreads();
  for (int off = 128; off > 0; off >>= 1) {
    if (t < off) sh[t] += sh[t + off];
    __syncthreads();
  }
  const float inv = 1.f / sh[0];
  if (t < HW) coef[base + (size_t)t * NCLS] = ex * inv;
}

// ---------------------------------------------------------------------------
// Kernel 6 (WMMA): out[b,c,d] = sum_hw coef[b,hw,c] * img[b,d,hw]
//   Per-batch GEMM (80x196)x(196x2048), K=196=49*4 WMMA steps, one wave per
//   16x16 output tile, direct (L2-resident) fragment loads.
// ---------------------------------------------------------------------------
__global__ __launch_bounds__(128) void k_pool(const float* __restrict__ coef,
                                              const float* __restrict__ img,
                                              float* __restrict__ out) {
  const int tid = threadIdx.x;
  const int lane = tid & 31;
  const int wid = tid >> 5;
  const int lmod = lane & 15, lhalf = lane >> 4;
  const int tile = blockIdx.x * 4 + wid;       // 0..5119
  const int b = tile / (5 * 128);
  const int rem = tile - b * 5 * 128;
  const int ct = rem / 128;                    // class tile  (80 / 16 = 5)
  const int dt = rem - ct * 128;               // d tile      (2048/16 = 128)
  const int c = ct * 16 + lmod;
  const int d = dt * 16 + lmod;
  const float* coefB = coef + (size_t)b * HW * NCLS;
  const float* imgB  = img  + (size_t)b * D_IMG * HW;
  v8f acc = {};
  for (int k0 = 0; k0 < HW; k0 += 4) {
    const int k = k0 + 2 * lhalf;
    v2f a, bb;
    a.x  = coefB[(size_t)k * NCLS + c];
    a.y  = coefB[(size_t)(k + 1) * NCLS + c];
    bb.x = imgB[(size_t)d * HW + k];
    bb.y = imgB[(size_t)d * HW + k + 1];
    acc = wmma_f32_4(a, bb, acc);
  }
#pragma unroll
  for (int i = 0; i < 8; ++i) {
    const int cc = ct * 16 + i + 8 * lhalf;
    out[((size_t)b * NCLS + cc) * D_IMG + d] = acc[i];
  }
}

// ---------------------------------------------------------------------------
extern "C" void kernel_launch(void* const* d_in, const int* in_sizes, int n_in,
                              void* d_out, int out_size, void* d_ws, size_t ws_size,
                              hipStream_t stream) {
  const float* img  = (const float*)d_in[1];  // [8,2048,14,14]
  const float* word = (const float*)d_in[2];  // [80,300]
  const float* W1   = (const float*)d_in[3];  // [1024,2048]
  const float* W2   = (const float*)d_in[4];  // [1024,300]
  const float* W3   = (const float*)d_in[5];  // [1024,1024]
  const float* b3   = (const float*)d_in[6];  // [1024]
  const float* Wa   = (const float*)d_in[7];  // [1,1024]
  const float* ba   = (const float*)d_in[8];  // [1]
  float* out = (float*)d_out;                 // [8,80,2048]

  float* ws   = (float*)d_ws;
  float* fwd_ = ws;                                  // 80*1024
  float* v    = fwd_ + (size_t)NCLS * D_MID;         // 1024
  float* c0   = v + D_MID;                           // 16 (padded)
  float* fwh  = c0 + 16;                             // 1568*1024
  float* coef = fwh + (size_t)NP * D_MID;            // 1568*80
  // total scratch: ~7.3 MB

  k_fwd     <<<(NCLS * D_MID + 255) / 256, 256, 0, stream>>>(word, W2, fwd_);
  k_vc      <<<5, 256, 0, stream>>>(W3, Wa, b3, ba, v, c0);
  k_gemm_fwh<<<dim3(32, 25), 256, 0, stream>>>(img, W1, fwh);
  k_coef    <<<NP, 256, 0, stream>>>(fwh, fwd_, v, c0, coef);
  k_softmax <<<NB * NCLS, 256, 0, stream>>>(coef);
  k_pool    <<<1280, 128, 0, stream>>>(coef, img, out);
}